// TPlanesEnc_59450937311360
// MI455X (gfx1250) — compile-verified
//
#include <hip/hip_runtime.h>
#include <stdint.h>

typedef float v4f __attribute__((ext_vector_type(4)));

#define TP_P 512
#define TP_F 32

// One thread = one (point, plane, 4-feature chunk) -> one float4 result.
// Output flat index: point*96 + plane*32 + chunk*4 == 4*g where
// g = point*24 + plane*8 + chunk  ==> lane-contiguous 512B/wave NT stores.
__global__ __launch_bounds__(256)
void tplanes_bilerp_kernel(const float* __restrict__ coords,
                           const float* __restrict__ tplanes,
                           float* __restrict__ out,
                           int npoints)
{
    __shared__ float sC[48];   // up to 12 points * 3 coords staged per block

    const int tid   = threadIdx.x;
    const int g0    = blockIdx.x * 256;
    const int total = npoints * 24;

    // ---- CDNA5 async gather: stage this block's coords into LDS ----
    const int p0 = g0 / 24;
    int pLast = (g0 + 255) / 24;
    if (pLast >= npoints) pLast = npoints - 1;
    const int nwords = (pLast - p0 + 1) * 3;      // <= 36 dwords
    if (tid < nwords) {
        unsigned ldsOff = (unsigned)(uintptr_t)&sC[tid];        // low 32 bits = LDS byte offset
        unsigned gOff   = (unsigned)((p0 * 3 + tid) * 4);       // byte offset into coords (<= 6.3MB)
        asm volatile("global_load_async_to_lds_b32 %0, %1, %2"
                     :: "v"(ldsOff), "v"(gOff), "s"(coords)
                     : "memory");
    }
    asm volatile("s_wait_asynccnt 0" ::: "memory");
    __syncthreads();

    const int g = g0 + tid;
    if (g >= total) return;

    const int point = g / 24;
    const int r     = g - point * 24;
    const int plane = r >> 3;     // 0..2
    const int c     = r & 7;      // feature chunk of 4

    const int li   = (point - p0) * 3;
    const float cx = sC[li + 0];
    const float cy = sC[li + 1];
    const float cz = sC[li + 2];

    // plane 0 samples (x,y); plane 1 samples (x,z); plane 2 samples (z,y)
    const float u = (plane == 2) ? cz : cx;
    const float v = (plane == 1) ? cz : cy;

    const float Pf = (float)TP_P;
    float x = fminf(fmaxf((u * 0.5f + 0.5f) * Pf - 0.5f, 0.0f), Pf - 1.0f);
    float y = fminf(fmaxf((v * 0.5f + 0.5f) * Pf - 0.5f, 0.0f), Pf - 1.0f);
    float x0f = floorf(x), y0f = floorf(y);
    int xi0 = (int)x0f, yi0 = (int)y0f;
    int xi1 = min(xi0 + 1, TP_P - 1);
    int yi1 = min(yi0 + 1, TP_P - 1);
    float fx = x - x0f, fy = y - y0f;
    float gx = 1.0f - fx, gy = 1.0f - fy;

    // texel gathers hit the L2-resident planes (96MB < 192MB L2)
    const float* pb = tplanes + (size_t)plane * (TP_P * TP_P * TP_F) + c * 4;
    const v4f f00 = *(const v4f*)(pb + (size_t)(yi0 * TP_P + xi0) * TP_F);
    const v4f f01 = *(const v4f*)(pb + (size_t)(yi0 * TP_P + xi1) * TP_F);
    const v4f f10 = *(const v4f*)(pb + (size_t)(yi1 * TP_P + xi0) * TP_F);
    const v4f f11 = *(const v4f*)(pb + (size_t)(yi1 * TP_P + xi1) * TP_F);

    v4f res = (f00 * gx + f01 * fx) * gy + (f10 * gx + f11 * fx) * fy;

    // streaming NT store: don't let the 201MB output stream evict the planes from L2
    __builtin_nontemporal_store(res, (v4f*)out + g);
}

extern "C" void kernel_launch(void* const* d_in, const int* in_sizes, int n_in,
                              void* d_out, int out_size, void* d_ws, size_t ws_size,
                              hipStream_t stream)
{
    const float* coords  = (const float*)d_in[0];   // [B, N, 3] fp32
    const float* tplanes = (const float*)d_in[1];   // [3, 512, 512, 32] fp32
    float* out = (float*)d_out;                     // [B, N, 96] fp32

    const int npoints = in_sizes[0] / 3;            // B*N = 524288
    const int total   = npoints * 24;               // one float4 per thread
    const int blocks  = (total + 255) / 256;        // 49152 blocks of 256

    tplanes_bilerp_kernel<<<blocks, 256, 0, stream>>>(coords, tplanes, out, npoints);
}